// GNN_75402445848811
// MI455X (gfx1250) — compile-verified
//
#include <hip/hip_runtime.h>
#include <math.h>

// Problem constants (match reference)
#define B_ 8
#define N_ 512
#define D_ 128
#define H_ 64
#define BN_ (B_ * N_)   // 4096

typedef float v2f __attribute__((ext_vector_type(2)));
typedef float v8f __attribute__((ext_vector_type(8)));
typedef int v4i_vs __attribute__((vector_size(16)));   // matches async-LDS builtin pointee

#if defined(__HIP_DEVICE_COMPILE__) && __has_builtin(__builtin_amdgcn_wmma_f32_16x16x4_f32)
#define HAVE_WMMA_F32 1
#else
#define HAVE_WMMA_F32 0
#endif

#if defined(__HIP_DEVICE_COMPILE__) &&                                    \
    __has_builtin(__builtin_amdgcn_global_load_async_to_lds_b128) &&      \
    __has_builtin(__builtin_amdgcn_s_wait_asynccnt)
#define HAVE_ASYNC_LDS 1
#define GPTR(p) ((__attribute__((address_space(1))) v4i_vs*)(p))
#define LPTR(p) ((__attribute__((address_space(3))) v4i_vs*)(p))
#endif

#define A_STRIDE 516   // 512 + 4 pad: 516 % 64 == 4 -> conflict-free fragment reads
#define X_PAD 4

__device__ __forceinline__ float sigm(float x) { return 1.0f / (1.0f + expf(-x)); }

__device__ __forceinline__ float wave_sum(float v) {
#pragma unroll
  for (int off = 16; off; off >>= 1) v += __shfl_xor(v, off, 32);
  return v;
}

// ---------------- workspace layout (float offsets) ----------------
// ws[0]=c1 (weight-MLP collapsed scalar), ws[1]=loss accumulator, ws[2]=sum(kappa)
static constexpr size_t OFF_KAP  = 16;                       // [4096]
static constexpr size_t OFF_RANK = OFF_KAP  + BN_;           // [4096]
static constexpr size_t OFF_DEG  = OFF_RANK + BN_;           // [4096] rowsum(A)
static constexpr size_t OFF_F    = OFF_DEG  + BN_;           // [4096*3]
static constexpr size_t OFF_MV1  = OFF_F    + BN_ * 3;       // [4096*6] A@f, A@f^2
static constexpr size_t OFF_GAM  = OFF_MV1  + BN_ * 6;       // [4096*3]
static constexpr size_t OFF_DEL  = OFF_GAM  + BN_ * 3;       // [4096*3]
static constexpr size_t OFF_FD   = OFF_DEL  + BN_ * 3;       // [4096*3]
static constexpr size_t OFF_MV2  = OFF_FD   + BN_ * 3;       // [4096*9]
static constexpr size_t OFF_S    = OFF_MV2  + BN_ * 9;       // [8*320]
static constexpr size_t OFF_HM   = 126976;                   // [4096*128] masked h
static constexpr size_t OFF_AGG  = OFF_HM   + BN_ * 128;     // [4096*128]
static constexpr size_t OFF_Z1   = OFF_AGG  + BN_ * 128;     // [4096*64]
static constexpr size_t OFF_H1   = OFF_Z1   + BN_ * 64;      // [4096*64]
static constexpr size_t OFF_H2   = OFF_H1   + BN_ * 64;
static constexpr size_t OFF_H3   = OFF_H2   + BN_ * 64;

// ------------------------------------------------------------------
// init: collapse the per-entry weight MLP to c1 = sigmoid(MLP(1));
// zero the atomic accumulators (ws is NOT re-poisoned between replays).
__global__ void k_init(const float* wmW1, const float* wmb1, const float* wmW2,
                       const float* wmb2, const float* wmW3, const float* wmb3,
                       float* ws) {
  if (threadIdx.x == 0) {
    float h1[64];
    for (int k = 0; k < 64; k++) h1[k] = fmaxf(wmW1[k] + wmb1[k], 0.0f);
    float o = wmb3[0];
    for (int j = 0; j < 32; j++) {
      float a = wmb2[j];
      for (int k = 0; k < 64; k++) a += h1[k] * wmW2[k * 32 + j];
      o += fmaxf(a, 0.0f) * wmW3[j];
    }
    ws[0] = sigm(o);
    ws[1] = 0.0f;  // curv loss accumulator
    ws[2] = 0.0f;  // sum(kappa)
  }
}

// ------------------------------------------------------------------
// Per-node MLPs: kappa + 3 curvature functions f_c = sigmoid(relu(x@W1+b1)@W2+b2).
__global__ void k_node_mlp(const float* __restrict__ X,
                           const float* cW1, const float* cb1, const float* cW2, const float* cb2,
                           const float* fW1, const float* fb1, const float* fW2, const float* fb2,
                           float* kap, float* f_all, float* kap_sum) {
  int node = blockIdx.x;
  __shared__ float xs[128];
  __shared__ float red[4][64];
  int tid = threadIdx.x;
  if (tid < 128) xs[tid] = X[node * 128 + tid];
  __syncthreads();
  int c = tid >> 6;    // 0=kappa, 1..3 = fn
  int hc = tid & 63;
  const float* W1 = (c == 0) ? cW1 : fW1 + (c - 1) * 128 * 64;
  const float* b1 = (c == 0) ? cb1 : fb1 + (c - 1) * 64;
  const float* W2 = (c == 0) ? cW2 : fW2 + (c - 1) * 64;
  float b2 = (c == 0) ? cb2[0] : fb2[c - 1];
  float acc = b1[hc];
  for (int k = 0; k < 128; k++) acc += xs[k] * W1[k * 64 + hc];
  red[c][hc] = fmaxf(acc, 0.0f) * W2[hc];
  __syncthreads();
  if (hc < 32) red[c][hc] += red[c][hc + 32];
  __syncthreads();
  if (hc < 16) red[c][hc] += red[c][hc + 16];
  __syncthreads();
  if (hc < 8)  red[c][hc] += red[c][hc + 8];
  __syncthreads();
  if (hc < 4)  red[c][hc] += red[c][hc + 4];
  __syncthreads();
  if (hc < 2)  red[c][hc] += red[c][hc + 2];
  __syncthreads();
  if (hc == 0) {
    float v = sigm(red[c][0] + red[c][1] + b2);
    if (c == 0) { kap[node] = v; atomicAdd(kap_sum, v); }
    else        { f_all[node * 3 + (c - 1)] = v; }
  }
}

// ------------------------------------------------------------------
// wave-per-row: rank (replicates lax.top_k nesting/tie-break) + row-degree of A.
__global__ void k_row_stats(const float* __restrict__ A, const float* __restrict__ kap,
                            float* rank, float* degA) {
  int row = blockIdx.x * (blockDim.x >> 5) + (threadIdx.x >> 5);
  if (row >= BN_) return;
  int lane = threadIdx.x & 31;
  int b = row >> 9, i = row & (N_ - 1);
  const float* kb = kap + b * N_;
  float ki = kb[i];
  int abase = row * N_;
  float r = 0.0f, s = 0.0f;
  for (int j = lane; j < N_; j += 32) {
    float kj = kb[j];
    r += ((kj > ki) || (kj == ki && j < i)) ? 1.0f : 0.0f;
    s += A[abase + j];
  }
  r = wave_sum(r);
  s = wave_sum(s);
  if (lane == 0) { rank[row] = r; degA[row] = s; }
}

// ------------------------------------------------------------------
// wave-per-row matvec pass 1: A@f_c and A@(f_c^2), c=0..2 (6 dots, one A sweep)
__global__ void k_mv1(const float* __restrict__ A, const float* __restrict__ f_all,
                      float* mv1) {
  int row = blockIdx.x * (blockDim.x >> 5) + (threadIdx.x >> 5);
  if (row >= BN_) return;
  int lane = threadIdx.x & 31;
  int fbase = (row >> 9) << 9;   // b*512
  int abase = row * N_;
  float acc[6] = {0, 0, 0, 0, 0, 0};
  for (int j = lane; j < N_; j += 32) {
    float a = A[abase + j];
    int jb = (fbase + j) * 3;
#pragma unroll
    for (int c = 0; c < 3; c++) {
      float fv = f_all[jb + c];
      acc[c]     += a * fv;
      acc[3 + c] += a * fv * fv;
    }
  }
#pragma unroll
  for (int q = 0; q < 6; q++) acc[q] = wave_sum(acc[q]);
  if (lane == 0) {
#pragma unroll
    for (int q = 0; q < 6; q++) mv1[row * 6 + q] = acc[q];
  }
}

// gamma / delta_f / f*delta_f (elementwise, weights = c1*A)
__global__ void k_gamma(const float* __restrict__ f_all, const float* __restrict__ mv1,
                        const float* __restrict__ degA, const float* ws,
                        float* gam, float* del, float* fd) {
  int idx = blockIdx.x * blockDim.x + threadIdx.x;
  if (idx >= BN_) return;
  float c1 = ws[0];
  float degv = c1 * degA[idx];
#pragma unroll
  for (int c = 0; c < 3; c++) {
    float f = f_all[idx * 3 + c];
    float wf = c1 * mv1[idx * 6 + c];
    float wff = c1 * mv1[idx * 6 + 3 + c];
    float g = 0.5f * (f * f * degv - 2.0f * f * wf + wff);
    float d = f * degv - wf;
    gam[idx * 3 + c] = g;
    del[idx * 3 + c] = d;
    fd[idx * 3 + c] = f * d;
  }
}

// wave-per-row matvec pass 2: A@gamma, A@delta, A@(f*delta) (9 dots, one A sweep)
__global__ void k_mv2(const float* __restrict__ A, const float* __restrict__ gam,
                      const float* __restrict__ del, const float* __restrict__ fd,
                      float* mv2) {
  int row = blockIdx.x * (blockDim.x >> 5) + (threadIdx.x >> 5);
  if (row >= BN_) return;
  int lane = threadIdx.x & 31;
  int fbase = (row >> 9) << 9;
  int abase = row * N_;
  float acc[9] = {0, 0, 0, 0, 0, 0, 0, 0, 0};
  for (int j = lane; j < N_; j += 32) {
    float a = A[abase + j];
    int jb = (fbase + j) * 3;
#pragma unroll
    for (int c = 0; c < 3; c++) {
      acc[c]     += a * gam[jb + c];
      acc[3 + c] += a * del[jb + c];
      acc[6 + c] += a * fd[jb + c];
    }
  }
#pragma unroll
  for (int q = 0; q < 9; q++) acc[q] = wave_sum(acc[q]);
  if (lane == 0) {
#pragma unroll
    for (int q = 0; q < 9; q++) mv2[row * 9 + q] = acc[q];
  }
}

// loss: sum relu(kap*gamma - gamma2), wave pre-reduced before atomics
__global__ void k_loss(const float* __restrict__ f_all, const float* __restrict__ gam,
                       const float* __restrict__ del, const float* __restrict__ mv1,
                       const float* __restrict__ mv2, const float* __restrict__ degA,
                       const float* __restrict__ kap, float* ws) {
  int idx = blockIdx.x * blockDim.x + threadIdx.x;
  float local = 0.0f;
  if (idx < BN_) {
    float c1 = ws[0];
    float degv = c1 * degA[idx];
    float kv = kap[idx];
#pragma unroll
    for (int c = 0; c < 3; c++) {
      float f = f_all[idx * 3 + c];
      float g = gam[idx * 3 + c];
      float d = del[idx * 3 + c];
      float wf = c1 * mv1[idx * 6 + c];
      float Ag = mv2[idx * 9 + c];
      float Ad = mv2[idx * 9 + 3 + c];
      float Afd = mv2[idx * 9 + 6 + c];
      float delta_gamma = g * degv - c1 * Ag;
      float gfd = 0.5f * (f * d * degv - f * (c1 * Ad) - d * wf + c1 * Afd);
      float gamma2 = 0.5f * delta_gamma - gfd;
      local += fmaxf(kv * g - gamma2, 0.0f);
    }
  }
  local = wave_sum(local);
  if ((threadIdx.x & 31) == 0) atomicAdd(&ws[1], local);
}

// ------------------------------------------------------------------
// hm[node, k] = mask(node) * src[node, k]; mask from cumulative nested top-k
__global__ void k_mask_copy(const float* __restrict__ src, const float* __restrict__ rank,
                            const int* p_ptr, int t, float* dst, int K, int total) {
  int idx = blockIdx.x * blockDim.x + threadIdx.x;
  if (idx >= total) return;
  int node = idx / K;
  float num = (float)((N_ * p_ptr[0] * t) / 100);
  float m = (rank[node] >= num) ? 1.0f : 0.0f;
  dst[idx] = m * src[idx];
}

// ------------------------------------------------------------------
// agg[b] = (1+eps)*h[b] + m_i * (A[b] @ (m_j h[b]))
// One workgroup per (16-row tile, batch): the 16x512 A block (contiguous 32KB)
// is staged once into LDS (async-to-LDS), then K/16 waves each compute one
// 16x16 tile with fp32 WMMA; A-fragments via conflict-free ds_load_2addr_b64.
// All global indexing is 32-bit from wave-uniform bases -> SADDR+voffset form.
__global__ void k_spmm_wmma(const float* __restrict__ A, const float* __restrict__ Hm,
                            const float* __restrict__ Hin, const float* __restrict__ rank,
                            const float* eps_p, const int* p_ptr, int t,
                            float* agg, int K) {
  __shared__ float As[16 * A_STRIDE];
  int b = blockIdx.y;
  int tm = blockIdx.x << 4;
  int tid = threadIdx.x;
  int lane = tid & 31;
  int wave = tid >> 5;               // 0 .. K/16-1
  int half = lane >> 4, l15 = lane & 15;
  int aoff = (b * N_ + tm) * N_;     // 16 contiguous rows of A
  int hboff = b * N_ * K;            // batch base inside Hm
  int tn = wave << 4;
  __builtin_prefetch(Hm + hboff + tn, 0, 1);

  // cooperative stage of A tile (row-padded in LDS)
  for (int v = tid; v < 16 * 128; v += blockDim.x) {
    int row = v >> 7, seg = (v & 127) << 2;
#ifdef HAVE_ASYNC_LDS
    __builtin_amdgcn_global_load_async_to_lds_b128(
        GPTR(A + aoff + row * N_ + seg), LPTR(As + row * A_STRIDE + seg), 0, 0);
#else
    *(float4*)(As + row * A_STRIDE + seg) = *(const float4*)(A + aoff + row * N_ + seg);
#endif
  }
#ifdef HAVE_ASYNC_LDS
  __builtin_amdgcn_s_wait_asynccnt(0);
#endif
  __syncthreads();

  const float* Arow = As + l15 * A_STRIDE;
  int bofs = hboff + tn + l15 + half * 2 * K;   // per-lane, loop-invariant
  v8f c = {};
#if HAVE_WMMA_F32
  for (int k0 = 0; k0 < N_; k0 += 4) {
    int ka = k0 + half * 2;
    v2f av, bv;
    av.x = Arow[ka];
    av.y = Arow[ka + 1];
    bv.x = Hm[bofs + k0 * K];
    bv.y = Hm[bofs + k0 * K + K];
    c = __builtin_amdgcn_wmma_f32_16x16x4_f32(false, av, false, bv, (short)0, c,
                                              false, false);
  }
#else
  for (int k0 = 0; k0 < N_; k0++) {
    float bval = Hm[hboff + k0 * K + tn + l15];
#pragma unroll
    for (int r = 0; r < 8; r++)
      c[r] += As[(r + half * 8) * A_STRIDE + k0] * bval;
  }
#endif
  float eps1 = 1.0f + eps_p[0];
  float num = (float)((N_ * p_ptr[0] * t) / 100);
#pragma unroll
  for (int r = 0; r < 8; r++) {
    int row = tm + r + half * 8;
    int node = b * N_ + row;
    float m = (rank[node] >= num) ? 1.0f : 0.0f;
    int o = node * K + tn + l15;
    agg[o] = eps1 * Hin[o] + m * c[r];
  }
}

// out[4096,64] = relu(Xin[4096,K] @ W[K,64] + bias)
// One workgroup (4 waves) per 16-row tile; 16xK activation tile staged in LDS.
__global__ void k_mlp_wmma(const float* __restrict__ Xin, const float* __restrict__ W,
                           const float* __restrict__ bias, float* out, int K) {
  __shared__ float Xs[16 * (128 + X_PAD)];
  int stride = K + X_PAD;
  int tm = blockIdx.x << 4;
  int tid = threadIdx.x;
  int lane = tid & 31;
  int wave = tid >> 5;               // 0..3
  int half = lane >> 4, l15 = lane & 15;
  int xoff = tm * K;                 // 16 contiguous rows
  int kv4 = K >> 2;

  for (int v = tid; v < 16 * kv4; v += blockDim.x) {
    int row = v / kv4, seg = (v % kv4) << 2;
#ifdef HAVE_ASYNC_LDS
    __builtin_amdgcn_global_load_async_to_lds_b128(
        GPTR(Xin + xoff + row * K + seg), LPTR(Xs + row * stride + seg), 0, 0);
#else
    *(float4*)(Xs + row * stride + seg) = *(const float4*)(Xin + xoff + row * K + seg);
#endif
  }
#ifdef HAVE_ASYNC_LDS
  __builtin_amdgcn_s_wait_asynccnt(0);
#endif
  __syncthreads();

  int tn = wave << 4;
  const float* Arow = Xs + l15 * stride;
  int bofs = tn + l15 + half * 2 * 64;          // per-lane, loop-invariant
  v8f c = {};
#if HAVE_WMMA_F32
  for (int k0 = 0; k0 < K; k0 += 4) {
    int ka = k0 + half * 2;
    v2f av, bv;
    av.x = Arow[ka];
    av.y = Arow[ka + 1];
    bv.x = W[bofs + k0 * 64];
    bv.y = W[bofs + k0 * 64 + 64];
    c = __builtin_amdgcn_wmma_f32_16x16x4_f32(false, av, false, bv, (short)0, c,
                                              false, false);
  }
#else
  for (int k0 = 0; k0 < K; k0++) {
    float bval = W[k0 * 64 + tn + l15];
#pragma unroll
    for (int r = 0; r < 8; r++)
      c[r] += Xs[(r + half * 8) * stride + k0] * bval;
  }
#endif
#pragma unroll
  for (int r = 0; r < 8; r++) {
    int row = tm + r + half * 8;
    out[row * 64 + tn + l15] = fmaxf(c[r] + bias[tn + l15], 0.0f);
  }
}

// ------------------------------------------------------------------
// column-sum of concat(X,h1,h2,h3) over nodes -> s[8,320]
__global__ void k_colsum(const float* __restrict__ X, const float* __restrict__ h1,
                         const float* __restrict__ h2, const float* __restrict__ h3,
                         float* s) {
  int idx = blockIdx.x * blockDim.x + threadIdx.x;
  if (idx >= B_ * 320) return;
  int b = idx / 320, cf = idx % 320;
  const float* src;
  int K, c;
  if (cf < 128)      { src = X;  K = 128; c = cf; }
  else if (cf < 192) { src = h1; K = 64;  c = cf - 128; }
  else if (cf < 256) { src = h2; K = 64;  c = cf - 192; }
  else               { src = h3; K = 64;  c = cf - 256; }
  float acc = 0.0f;
  int base = b * N_ * K + c;
  for (int n = 0; n < N_; n++) acc += src[base + n * K];
  s[idx] = acc;
}

// final projection + scalar loss write
__global__ void k_out(const float* __restrict__ s, const float* __restrict__ outW,
                      const float* __restrict__ outb, const float* ws, float* dout) {
  int tid = threadIdx.x;
  if (tid < 80) {
    int b = tid / 10, o = tid % 10;
    float acc = outb[o];
    const float* sb = s + b * 320;
    for (int cc = 0; cc < 320; cc++) acc += sb[cc] * outW[cc * 10 + o];
    dout[tid] = acc;
  } else if (tid == 80) {
    dout[80] = ws[1] - 3.0f * ws[2];
  }
}

// ------------------------------------------------------------------
extern "C" void kernel_launch(void* const* d_in, const int* in_sizes, int n_in,
                              void* d_out, int out_size, void* d_ws, size_t ws_size,
                              hipStream_t stream) {
  (void)in_sizes; (void)n_in; (void)out_size; (void)ws_size;
  const float* X = (const float*)d_in[0];
  const float* A = (const float*)d_in[1];
  const int* p_ptr = (const int*)d_in[2];
  const float* curv_W1 = (const float*)d_in[3];
  const float* curv_b1 = (const float*)d_in[4];
  const float* curv_W2 = (const float*)d_in[5];
  const float* curv_b2 = (const float*)d_in[6];
  const float* wm_W1 = (const float*)d_in[7];
  const float* wm_b1 = (const float*)d_in[8];
  const float* wm_W2 = (const float*)d_in[9];
  const float* wm_b2 = (const float*)d_in[10];
  const float* wm_W3 = (const float*)d_in[11];
  const float* wm_b3 = (const float*)d_in[12];
  const float* fn_W1 = (const float*)d_in[13];
  const float* fn_b1 = (const float*)d_in[14];
  const float* fn_W2 = (const float*)d_in[15];
  const float* fn_b2 = (const float*)d_in[16];
  const float* gin_eps = (const float*)d_in[17];
  const float* ginW[3][4] = {
      {(const float*)d_in[18], (const float*)d_in[19], (const float*)d_in[20], (const float*)d_in[21]},
      {(const float*)d_in[22], (const float*)d_in[23], (const float*)d_in[24], (const float*)d_in[25]},
      {(const float*)d_in[26], (const float*)d_in[27], (const float*)d_in[28], (const float*)d_in[29]}};
  const float* out_W = (const float*)d_in[30];
  const float* out_b = (const float*)d_in[31];

  float* ws = (float*)d_ws;
  float* kap  = ws + OFF_KAP;
  float* rank = ws + OFF_RANK;
  float* degA = ws + OFF_DEG;
  float* f    = ws + OFF_F;
  float* mv1  = ws + OFF_MV1;
  float* gam  = ws + OFF_GAM;
  float* del  = ws + OFF_DEL;
  float* fd   = ws + OFF_FD;
  float* mv2  = ws + OFF_MV2;
  float* s    = ws + OFF_S;
  float* hm   = ws + OFF_HM;
  float* agg  = ws + OFF_AGG;
  float* z1   = ws + OFF_Z1;
  float* hbuf[3] = {ws + OFF_H1, ws + OFF_H2, ws + OFF_H3};
  float* dout = (float*)d_out;

  const int T = 256;
  const int waveRowBlocks = BN_ / (T / 32);   // 512 blocks, wave-per-row kernels
  const int rowBlocks = (BN_ + T - 1) / T;

  k_init<<<1, 32, 0, stream>>>(wm_W1, wm_b1, wm_W2, wm_b2, wm_W3, wm_b3, ws);
  k_node_mlp<<<BN_, 256, 0, stream>>>(X, curv_W1, curv_b1, curv_W2, curv_b2,
                                      fn_W1, fn_b1, fn_W2, fn_b2, kap, f, &ws[2]);
  k_row_stats<<<waveRowBlocks, T, 0, stream>>>(A, kap, rank, degA);
  k_mv1<<<waveRowBlocks, T, 0, stream>>>(A, f, mv1);
  k_gamma<<<rowBlocks, T, 0, stream>>>(f, mv1, degA, ws, gam, del, fd);
  k_mv2<<<waveRowBlocks, T, 0, stream>>>(A, gam, del, fd, mv2);
  k_loss<<<rowBlocks, T, 0, stream>>>(f, gam, del, mv1, mv2, degA, kap, ws);

  // GIN layers
  for (int t = 0; t < 3; t++) {
    int K = (t == 0) ? 128 : 64;
    const float* hin = (t == 0) ? X : hbuf[t - 1];
    float* hout = hbuf[t];
    int total = BN_ * K;
    k_mask_copy<<<(total + T - 1) / T, T, 0, stream>>>(hin, rank, p_ptr, t, hm, K, total);
    dim3 g(N_ / 16, B_);                 // 32 row tiles x 8 batches
    k_spmm_wmma<<<g, 32 * (K / 16), 0, stream>>>(A, hm, hin, rank, gin_eps + t,
                                                 p_ptr, t, agg, K);
    k_mlp_wmma<<<BN_ / 16, 128, 0, stream>>>(agg, ginW[t][0], ginW[t][1], z1, K);
    k_mlp_wmma<<<BN_ / 16, 128, 0, stream>>>(z1, ginW[t][2], ginW[t][3], hout, 64);
  }

  k_colsum<<<(B_ * 320 + T - 1) / T, T, 0, stream>>>(X, hbuf[0], hbuf[1], hbuf[2], s);
  k_out<<<1, 96, 0, stream>>>(s, out_W, out_b, ws, dout);
}